// MultiTierLoss_86466281603355
// MI455X (gfx1250) — compile-verified
//
#include <hip/hip_runtime.h>
#include <hip/hip_bf16.h>

#define D_COLS   1000
#define B_ROWS   4096
#define T0_S1_HI 50
#define T0_S2_LO 475
#define T1_S1_LO 475
#define T1_S2_LO 950
#define MAX_REL_ 8

typedef float v2f __attribute__((ext_vector_type(2)));
typedef float v8f __attribute__((ext_vector_type(8)));

// One block per row: stage the 1000-float row in LDS, evaluate all hinge
// pairs with broadcast-friendly access (all lanes read the same s[i],
// strided s[j]), reduce within the block, write per-row loss to d_ws.
__global__ __launch_bounds__(256)
void MultiTierLoss_row_kernel(const float* __restrict__ scores,
                              const int* __restrict__ labels,
                              float* __restrict__ row_loss) {
  __shared__ float s[D_COLS];
  __shared__ float wpart[8];
  __shared__ int   sn;

  const int row = blockIdx.x;
  const int tid = threadIdx.x;
  const float* __restrict__ srow = scores + (size_t)row * D_COLS;

  for (int c = tid; c < D_COLS; c += 256) s[c] = srow[c];
  if (tid == 0) {
    const int* __restrict__ lrow = labels + (size_t)row * D_COLS;
    int n = 0;
    for (int c = 0; c < MAX_REL_; ++c) n += (lrow[c] > -1) ? 1 : 0;
    sn = n;
  }
  __syncthreads();
  const int n = sn;

  float acc0 = 0.0f, acc1 = 0.0f, accg = 0.0f;

  // Tier 0: i in [0,50), j in [475,1000)  -> 26250 pairs
  for (int i = 0; i < T0_S1_HI; ++i) {
    const float c = 1.0f - s[i];
    for (int j = T0_S2_LO + tid; j < D_COLS; j += 256)
      acc0 += fmaxf(0.0f, c + s[j]);
  }

  // Tier 1: i in [475,525), j in [950,1000) -> 2500 pairs (flat index)
  for (int p = tid; p < 2500; p += 256) {
    const int i = T1_S1_LO + p / 50;
    const int j = T1_S2_LO + p % 50;
    acc1 += fmaxf(0.0f, 1.0f - s[i] + s[j]);
  }

  // GT: i in [0,n), j in [n,1000)
  for (int i = 0; i < n; ++i) {
    const float c = 1.0f - s[i];
    for (int j = n + tid; j < D_COLS; j += 256)
      accg += fmaxf(0.0f, c + s[j]);
  }

  float v = acc0 * (1.0f / (50.0f * 525.0f))
          + acc1 * (1.0f / 2500.0f)
          + accg * (2.0f / (float)D_COLS);

  // wave32 shuffle reduction, then cross-wave via LDS
  for (int off = 16; off > 0; off >>= 1) v += __shfl_down(v, off, 32);
  const int lane = tid & 31, wid = tid >> 5;
  if (lane == 0) wpart[wid] = v;
  __syncthreads();
  if (tid == 0) {
    float t = 0.0f;
    for (int w = 0; w < 8; ++w) t += wpart[w];
    row_loss[row] = t;
  }
}

// Single wave32: sum 4096 floats via chained V_WMMA_F32_16X16X4_F32 with an
// all-ones B. D[m][n] += sum_k A[m][k], so Sum(all D) = 16 * Sum(all A) for
// ANY bijective placement of the 64-value chunk into A slots. Deterministic.
__global__ __launch_bounds__(32)
void MultiTierLoss_wmma_mean_kernel(const float* __restrict__ row_loss,
                                    float* __restrict__ out) {
  const int lane = threadIdx.x;       // 0..31, EXEC all ones (WMMA requirement)
  const int m    = lane & 15;
  const int kb   = (lane >> 4) * 2;   // lanes 0-15 -> K{0,1}, lanes 16-31 -> K{2,3}

  v2f bones; bones.x = 1.0f; bones.y = 1.0f;
  v8f c;
  #pragma unroll
  for (int i = 0; i < 8; ++i) c[i] = 0.0f;

  for (int chunk = 0; chunk < B_ROWS; chunk += 64) {
    v2f a;
    a.x = row_loss[chunk + m * 4 + kb + 0];
    a.y = row_loss[chunk + m * 4 + kb + 1];
    // 8 args: (neg_a, A, neg_b, B, c_mod, C, reuse_a, reuse_b)
    c = __builtin_amdgcn_wmma_f32_16x16x4_f32(false, a, false, bones,
                                              (short)0, c, false, false);
  }

  float ssum = 0.0f;
  #pragma unroll
  for (int i = 0; i < 8; ++i) ssum += c[i];
  for (int off = 16; off > 0; off >>= 1) ssum += __shfl_down(ssum, off, 32);

  if (lane == 0)
    out[0] = ssum * (1.0f / (16.0f * (float)B_ROWS));  // /16 dup columns, /B mean
}

extern "C" void kernel_launch(void* const* d_in, const int* in_sizes, int n_in,
                              void* d_out, int out_size, void* d_ws, size_t ws_size,
                              hipStream_t stream) {
  (void)in_sizes; (void)n_in; (void)out_size; (void)ws_size;
  const float* scores = (const float*)d_in[0];
  const int*   labels = (const int*)d_in[1];
  float* rl = (float*)d_ws;                 // 4096 floats of scratch

  MultiTierLoss_row_kernel<<<B_ROWS, 256, 0, stream>>>(scores, labels, rl);
  MultiTierLoss_wmma_mean_kernel<<<1, 32, 0, stream>>>(rl, (float*)d_out);
}